// PrototypicalLoss_14370960572756
// MI455X (gfx1250) — compile-verified
//
#include <hip/hip_runtime.h>
#include <hip/hip_bf16.h>
#include <math.h>

// PrototypicalLoss for MI455X (gfx1250, wave32).
// B=8192, N=64, D=256. Memory-bound: ~522MB @ 23.3TB/s => ~22us floor.
// Per-b workgroup streams support tile into LDS (coalesced b128), does the
// masked-sum einsum with V_WMMA_F32_16X16X4_F32 (A = mask broadcast over M),
// then distance + deterministic tree reductions. No float atomics.

typedef float v2f __attribute__((ext_vector_type(2)));
typedef float v8f __attribute__((ext_vector_type(8)));

#define N_SUP 64
#define DIM 256
#define LDS_STRIDE 272   // 256 + 16 pad: row r shifts banks by 16 -> half-waves conflict-free
#define CHUNK_ROWS 32

__global__ void __launch_bounds__(256, 1)
proto_dist_kernel(const float* __restrict__ query,
                  const float* __restrict__ support,
                  const int* __restrict__ labels,
                  float* __restrict__ dist)
{
    __shared__ float sTile[CHUNK_ROWS * LDS_STRIDE]; // 34816 B
    __shared__ float sMask[N_SUP];
    __shared__ float sProto[DIM];
    __shared__ float sRed[256];
    __shared__ float sCount;

    const int b    = blockIdx.x;
    const int t    = threadIdx.x;
    const int lane = t & 31;
    const int wave = t >> 5;
    const bool hi  = lane >= 16;
    const int colLane = lane & 15;

    const float* sup = support + (size_t)b * (N_SUP * DIM);

    // mask + count
    if (t < N_SUP) sMask[t] = (labels[b * N_SUP + t] == 1) ? 1.0f : 0.0f;
    __syncthreads();
    if (t == 0) {
        float c = 0.0f;
        #pragma unroll
        for (int i = 0; i < N_SUP; ++i) c += sMask[i];
        sCount = c;
    }

    // each wave owns D-tiles {wave, wave+8}
    const int d0 = wave * 16;
    const int d1 = (wave + 8) * 16;
    v8f c0 = {}; v8f c1 = {};

    for (int ch = 0; ch < 2; ++ch) {
        __syncthreads(); // prev-chunk WMMA reads done (and mask/count ready)

        // stage 32 rows x 256 cols into LDS, coalesced float4
        const float4* g4 = (const float4*)(sup + ch * CHUNK_ROWS * DIM);
        #pragma unroll
        for (int i = 0; i < 8; ++i) {
            int L  = i * 256 + t;       // float4 linear index
            int n  = L >> 6;            // 64 float4 per row
            int d4 = L & 63;
            float4 v = g4[L];
            *(float4*)&sTile[n * LDS_STRIDE + d4 * 4] = v;
        }
        __syncthreads();

        #pragma unroll
        for (int kt = 0; kt < 8; ++kt) {
            // A (16x4 f32): VGPR0 = K0 (lanes0-15) / K2 (lanes16-31), VGPR1 = K1 / K3.
            // All M rows identical (broadcast mask) -> every D row equals proto chunk.
            const int kb = ch * CHUNK_ROWS + kt * 4 + (hi ? 2 : 0);
            const int rl = kt * 4 + (hi ? 2 : 0);
#if __has_builtin(__builtin_amdgcn_wmma_f32_16x16x4_f32)
            v2f a;  a.x = sMask[kb];  a.y = sMask[kb + 1];
            v2f b0; b0.x = sTile[rl * LDS_STRIDE + d0 + colLane];
                    b0.y = sTile[(rl + 1) * LDS_STRIDE + d0 + colLane];
            v2f b1; b1.x = sTile[rl * LDS_STRIDE + d1 + colLane];
                    b1.y = sTile[(rl + 1) * LDS_STRIDE + d1 + colLane];
            c0 = __builtin_amdgcn_wmma_f32_16x16x4_f32(false, a, false, b0,
                                                       (short)0, c0, false, false);
            c1 = __builtin_amdgcn_wmma_f32_16x16x4_f32(false, a, false, b1,
                                                       (short)0, c1, false, false);
#else
            // VALU fallback (only lanes 0-15 values consumed at write-out)
            const int kbase = ch * CHUNK_ROWS + kt * 4;
            const int rbase = kt * 4;
            float a0 = 0.0f, a1 = 0.0f;
            #pragma unroll
            for (int k = 0; k < 4; ++k) {
                float m = sMask[kbase + k];
                a0 += m * sTile[(rbase + k) * LDS_STRIDE + d0 + colLane];
                a1 += m * sTile[(rbase + k) * LDS_STRIDE + d1 + colLane];
            }
            c0[0] += a0;
            c1[0] += a1;
#endif
        }
    }

    // D row M=0 lives in VGPR0 of C, lanes 0-15
    if (lane < 16) {
        sProto[d0 + lane] = c0[0];
        sProto[d1 + lane] = c1[0];
    }
    __syncthreads();

    // distance: each thread owns one d
    const float q   = query[(size_t)b * DIM + t];
    const float fb  = sup[t];               // support[b,0,t] fallback (L2-hot)
    const float cnt = sCount;
    const float proto = (cnt > 0.5f) ? (sProto[t] / cnt) : fb;
    const float diff  = q - proto;
    sRed[t] = diff * diff;
    __syncthreads();
    #pragma unroll
    for (int s = 128; s > 0; s >>= 1) {
        if (t < s) sRed[t] += sRed[t + s];
        __syncthreads();
    }
    if (t == 0) dist[b] = sqrtf(sRed[0] + 1e-8f);
}

__global__ void __launch_bounds__(256, 1)
mean_kernel(const float* __restrict__ dist, float* __restrict__ out, int n)
{
    __shared__ float sRed[256];
    const int t = threadIdx.x;
    float acc = 0.0f;
    for (int i = t; i < n; i += 256) acc += dist[i];  // fixed order: deterministic
    sRed[t] = acc;
    __syncthreads();
    #pragma unroll
    for (int s = 128; s > 0; s >>= 1) {
        if (t < s) sRed[t] += sRed[t + s];
        __syncthreads();
    }
    if (t == 0) out[0] = sRed[0] / (float)n;
}

extern "C" void kernel_launch(void* const* d_in, const int* in_sizes, int n_in,
                              void* d_out, int out_size, void* d_ws, size_t ws_size,
                              hipStream_t stream)
{
    const float* query   = (const float*)d_in[0];  // (B, 256)
    const float* support = (const float*)d_in[1];  // (B, 64, 256)
    const int*   labels  = (const int*)d_in[2];    // (B, 64)
    float* out  = (float*)d_out;
    float* dist = (float*)d_ws;                    // B floats scratch

    const int B = in_sizes[0] / DIM;               // 8192

    proto_dist_kernel<<<B, 256, 0, stream>>>(query, support, labels, dist);
    mean_kernel<<<1, 256, 0, stream>>>(dist, out, B);
}